// CountVectorizer_35510789604072
// MI455X (gfx1250) — compile-verified
//
#include <hip/hip_runtime.h>

typedef __attribute__((ext_vector_type(2))) float v2f;
typedef __attribute__((ext_vector_type(8))) float v8f;

#define BATCH  1024
#define SEQ    200
#define DMODEL 128

// Embedding-bag reformulation: out[s,d] = sum_t W[tok[s,t], d] + bias[d].
// One wave handles one (sample, 16-dim tile); 8 waves per sample.
// The 4-token reduction is done by V_WMMA_F32_16X16X4_F32 with A = all-ones:
//   D[m,n] = sum_{k=0..3} B[k,n],  B[k, n=lane%16] = W[tok[t+k]][16*tile + n]
// B VGPR layout (f32 16x16x4): VGPR0 holds K = 0 (lanes 0-15) / 2 (lanes 16-31),
// VGPR1 holds K+1. Accumulation chains through C across 50 WMMAs.
__global__ __launch_bounds__(256) void countvec_embedbag_wmma(
    const int*   __restrict__ tok_ids,   // [BATCH, SEQ]
    const float* __restrict__ W,         // [VOCAB, DMODEL]
    const float* __restrict__ bias,      // [DMODEL]
    float*       __restrict__ out)       // [BATCH, DMODEL]
{
    const int lane    = threadIdx.x & 31;
    const int gwave   = blockIdx.x * 8 + (threadIdx.x >> 5);  // 8192 waves
    const int sample  = gwave >> 3;                           // 0..1023
    const int tile    = gwave & 7;                            // 0..7 -> dims [16*tile, +16)
    const int col     = lane & 15;                            // N within tile
    const bool lowHalf = lane < 16;                           // K base 0 vs 2

    // Broadcast token fetch: all lanes read the same int4 -> one b128 load/iter.
    const int4*  tok4 = (const int4*)(tok_ids + sample * SEQ);
    const float* Wc   = W + tile * 16 + col;   // column base for this tile

    const int d = tile * 16 + col;
    const float bsum = bias[d];                // issued early, overlaps gathers

    v8f c = {};
    v2f a;
    a.x = 1.0f;   // all-ones A: matrix pipe acts as a 4-way K reduction
    a.y = 1.0f;

#pragma unroll 10
    for (int i = 0; i < SEQ / 4; ++i) {
        const int4 tt = tok4[i];
        // Half-wave K split: lanes 0-15 take tokens {4i,4i+1}, lanes 16-31 {4i+2,4i+3}.
        const unsigned ta = (unsigned)(lowHalf ? tt.x : tt.z);
        const unsigned tb = (unsigned)(lowHalf ? tt.y : tt.w);

        v2f b;
        b.x = Wc[ta << 7];   // B[k0  ][col] : 64B contiguous burst per half-wave
        b.y = Wc[tb << 7];   // B[k0+1][col]

        // (neg_a, A, neg_b, B, c_mod, C, reuse_a, reuse_b)
        c = __builtin_amdgcn_wmma_f32_16x16x4_f32(
                false, a, false, b, (short)0, c, false, false);
    }

    // All D rows are identical (ones-A); lanes 0-15's c[0] holds the sum for
    // N = lane of this tile. Divergence only after the final WMMA.
    if (lowHalf) {
        out[sample * DMODEL + d] = c[0] + bsum;
    }
}

extern "C" void kernel_launch(void* const* d_in, const int* in_sizes, int n_in,
                              void* d_out, int out_size, void* d_ws, size_t ws_size,
                              hipStream_t stream) {
    (void)in_sizes; (void)n_in; (void)out_size; (void)d_ws; (void)ws_size;
    const int*   tok = (const int*)  d_in[0];   // token_ids [1024, 200] int32
    const float* W   = (const float*)d_in[1];   // [100000, 128] f32
    const float* b   = (const float*)d_in[2];   // [128] f32
    float*       out = (float*)d_out;           // [1024, 128] f32

    // 8192 waves = 1024 samples x 8 dim-tiles; 8 waves per 256-thread block.
    dim3 grid(BATCH * 8 / 8);
    dim3 block(256);
    hipLaunchKernelGGL(countvec_embedbag_wmma, grid, block, 0, stream,
                       tok, W, b, out);
}